// GCN_91096256348387
// MI455X (gfx1250) — compile-verified
//
#include <hip/hip_runtime.h>
#include <hip/hip_bf16.h>

typedef __attribute__((ext_vector_type(2))) float v2f;
typedef __attribute__((ext_vector_type(8))) float v8f;

// ---------------------------------------------------------------------------
// Degree / normalization
// ---------------------------------------------------------------------------
__global__ void deg_init_kernel(float* __restrict__ deg, int N) {
    int i = blockIdx.x * blockDim.x + threadIdx.x;
    if (i < N) deg[i] = 1.0f;               // self-loop contributes 1
}

__global__ void deg_count_kernel(const int* __restrict__ dst, float* __restrict__ deg, int E) {
    int e = blockIdx.x * blockDim.x + threadIdx.x;
    if (e < E) atomicAdd(&deg[dst[e]], 1.0f);
}

__global__ void dinv_kernel(float* __restrict__ deg_dinv, int N) {
    int i = blockIdx.x * blockDim.x + threadIdx.x;
    if (i < N) deg_dinv[i] = rsqrtf(deg_dinv[i]);   // deg >= 1 always
}

// ---------------------------------------------------------------------------
// Layer-1 GEMM: h[N,64] = x[N,128] @ W1[128,64] via V_WMMA_F32_16X16X4_F32.
// One wave per 16-row tile, covering all 4 column tiles (A-fragment reuse).
// Tile-count guard is wave-uniform so EXEC is all-1s at every WMMA.
// ---------------------------------------------------------------------------
__global__ __launch_bounds__(256) void gemm1_wmma_kernel(
        const float* __restrict__ x, const float* __restrict__ W1,
        float* __restrict__ h, int M /* rows, multiple of 16 */) {
    const int lane    = threadIdx.x & 31;
    const int waveId  = (blockIdx.x * (blockDim.x >> 5)) + (threadIdx.x >> 5);
    const int numTile = M >> 4;
    if (waveId >= numTile) return;          // wave-uniform exit

    const int row0    = waveId << 4;
    const int halfSel = lane >> 4;          // 0: lanes 0-15, 1: lanes 16-31
    const int l16     = lane & 15;

    v8f acc0 = {}, acc1 = {}, acc2 = {}, acc3 = {};

    const float* arow = x + (size_t)(row0 + l16) * 128;
    const int    kOff = halfSel ? 2 : 0;    // ISA: lanes16-31 hold K=k+2,k+3

    for (int k = 0; k < 128; k += 4) {
        v2f a;
        a.x = arow[k + kOff];
        a.y = arow[k + kOff + 1];

        const float* brow = W1 + (size_t)(k + kOff) * 64 + l16;
        v2f b0, b1, b2, b3;
        b0.x = brow[0];   b0.y = brow[64];
        b1.x = brow[16];  b1.y = brow[64 + 16];
        b2.x = brow[32];  b2.y = brow[64 + 32];
        b3.x = brow[48];  b3.y = brow[64 + 48];

        acc0 = __builtin_amdgcn_wmma_f32_16x16x4_f32(false, a, false, b0, (short)0, acc0, false, false);
        acc1 = __builtin_amdgcn_wmma_f32_16x16x4_f32(false, a, false, b1, (short)0, acc1, false, false);
        acc2 = __builtin_amdgcn_wmma_f32_16x16x4_f32(false, a, false, b2, (short)0, acc2, false, false);
        acc3 = __builtin_amdgcn_wmma_f32_16x16x4_f32(false, a, false, b3, (short)0, acc3, false, false);
    }

    // C/D layout: VGPR v, lanes 0-15 -> M=v, lanes 16-31 -> M=v+8; N = l16.
    float* hrow = h + (size_t)(row0 + halfSel * 8) * 64 + l16;
#pragma unroll
    for (int v = 0; v < 8; ++v) {
        hrow[(size_t)v * 64 + 0]  = acc0[v];
        hrow[(size_t)v * 64 + 16] = acc1[v];
        hrow[(size_t)v * 64 + 32] = acc2[v];
        hrow[(size_t)v * 64 + 48] = acc3[v];
    }
}

// ---------------------------------------------------------------------------
// out1[i,:] = b1[:] + dinv[i]^2 * h[i,:]   (bias + self-loop term)
// ---------------------------------------------------------------------------
__global__ void init_out1_kernel(const float* __restrict__ h, const float* __restrict__ dinv,
                                 const float* __restrict__ b1, float* __restrict__ out1, int N) {
    int t = blockIdx.x * blockDim.x + threadIdx.x;
    if (t >= N * 64) return;
    int i = t >> 6, j = t & 63;
    float di = dinv[i];
    out1[t] = b1[j] + di * di * h[t];
}

// ---------------------------------------------------------------------------
// Edge scatter, layer 1: out1[dst,:] += norm(e) * h[src,:]
// 16 lanes per edge, float4 gather, 4 f32 atomics each. L2-resident.
// ---------------------------------------------------------------------------
__global__ void scatter1_kernel(const float4* __restrict__ h4, const int* __restrict__ src,
                                const int* __restrict__ dst, const float* __restrict__ dinv,
                                float* __restrict__ out1, int E) {
    int t = blockIdx.x * blockDim.x + threadIdx.x;
    if (t >= E * 16) return;
    int e = t >> 4, q = t & 15;
    int s = src[e], d = dst[e];
    float nrm = dinv[s] * dinv[d];
    float4 v = h4[(size_t)s * 16 + q];
    float* o = out1 + (size_t)d * 64 + q * 4;
    atomicAdd(o + 0, nrm * v.x);
    atomicAdd(o + 1, nrm * v.y);
    atomicAdd(o + 2, nrm * v.z);
    atomicAdd(o + 3, nrm * v.w);
}

// ---------------------------------------------------------------------------
// Layer-2 GEMV with fused ReLU: g[i] = sum_j relu(out1[i,j]) * W2[j]
// One wave (32 lanes) per node; butterfly reduce.
// ---------------------------------------------------------------------------
__global__ void gemv2_relu_kernel(const float* __restrict__ out1, const float* __restrict__ W2,
                                  float* __restrict__ g, int N) {
    int wave = (blockIdx.x * blockDim.x + threadIdx.x) >> 5;
    int lane = threadIdx.x & 31;
    if (wave >= N) return;                  // wave-uniform
    const float* row = out1 + (size_t)wave * 64;
    float s = fmaxf(row[lane], 0.0f) * W2[lane]
            + fmaxf(row[lane + 32], 0.0f) * W2[lane + 32];
#pragma unroll
    for (int off = 16; off > 0; off >>= 1) s += __shfl_xor(s, off, 32);
    if (lane == 0) g[wave] = s;
}

// ---------------------------------------------------------------------------
// out2[i] = b2 + dinv[i]^2 * g[i]   (bias + self-loop), then edge scatter.
// ---------------------------------------------------------------------------
__global__ void init_out2_kernel(const float* __restrict__ g, const float* __restrict__ dinv,
                                 const float* __restrict__ b2, float* __restrict__ out2, int N) {
    int i = blockIdx.x * blockDim.x + threadIdx.x;
    if (i >= N) return;
    float di = dinv[i];
    out2[i] = b2[0] + di * di * g[i];
}

__global__ void scatter2_kernel(const float* __restrict__ g, const int* __restrict__ src,
                                const int* __restrict__ dst, const float* __restrict__ dinv,
                                float* __restrict__ out2, int E) {
    int e = blockIdx.x * blockDim.x + threadIdx.x;
    if (e >= E) return;
    int s = src[e], d = dst[e];
    atomicAdd(&out2[d], dinv[s] * dinv[d] * g[s]);
}

// ---------------------------------------------------------------------------
// Launcher
// ---------------------------------------------------------------------------
extern "C" void kernel_launch(void* const* d_in, const int* in_sizes, int n_in,
                              void* d_out, int out_size, void* d_ws, size_t ws_size,
                              hipStream_t stream) {
    const float* x   = (const float*)d_in[0];   // [N,128]
    const int*   ei  = (const int*)d_in[1];     // [2,E] flat: src then dst
    const float* W1  = (const float*)d_in[2];   // [128,64]
    const float* b1  = (const float*)d_in[3];   // [64]
    const float* W2  = (const float*)d_in[4];   // [64,1]
    const float* b2  = (const float*)d_in[5];   // [1]
    float*       out = (float*)d_out;           // [N]

    const int N = in_sizes[0] / 128;
    const int E = in_sizes[1] / 2;
    const int* src = ei;
    const int* dst = ei + E;

    // Workspace carve (all 256B-aligned chunks)
    char* ws = (char*)d_ws;
    size_t off = 0;
    auto carve = [&](size_t bytes) {
        void* p = ws + off;
        off += (bytes + 255) & ~(size_t)255;
        return p;
    };
    float* dinv = (float*)carve((size_t)N * sizeof(float));        // deg -> dinv in place
    float* h    = (float*)carve((size_t)N * 64 * sizeof(float));   // x @ W1
    float* out1 = (float*)carve((size_t)N * 64 * sizeof(float));   // layer-1 output
    float* g    = (float*)carve((size_t)N * sizeof(float));        // relu(out1) @ W2
    (void)ws_size;

    const int B = 256;
    auto cdiv = [](long long a, long long b) { return (int)((a + b - 1) / b); };

    // 1) degrees -> dinv
    deg_init_kernel<<<cdiv(N, B), B, 0, stream>>>(dinv, N);
    deg_count_kernel<<<cdiv(E, B), B, 0, stream>>>(dst, dinv, E);
    dinv_kernel<<<cdiv(N, B), B, 0, stream>>>(dinv, N);

    // 2) h = x @ W1  (WMMA f32 16x16x4; one wave per 16-row tile)
    {
        int tiles = N >> 4;                       // N = 50000 -> 3125 tiles exactly
        int wavesPerBlock = B >> 5;               // 8
        gemm1_wmma_kernel<<<cdiv(tiles, wavesPerBlock), B, 0, stream>>>(x, W1, h, N);
    }

    // 3) out1 = b1 + dinv^2 * h, then scatter edges
    init_out1_kernel<<<cdiv((long long)N * 64, B), B, 0, stream>>>(h, dinv, b1, out1, N);
    scatter1_kernel<<<cdiv((long long)E * 16, B), B, 0, stream>>>(
        (const float4*)h, src, dst, dinv, out1, E);

    // 4) g = relu(out1) @ W2   (wave per node)
    gemv2_relu_kernel<<<cdiv((long long)N * 32, B), B, 0, stream>>>(out1, W2, g, N);

    // 5) out = b2 + dinv^2 * g, then scatter edges
    init_out2_kernel<<<cdiv(N, B), B, 0, stream>>>(g, dinv, b2, out, N);
    scatter2_kernel<<<cdiv(E, B), B, 0, stream>>>(g, src, dst, dinv, out, E);
}